// CompositeLoss_6665789243961
// MI455X (gfx1250) — compile-verified
//
#include <hip/hip_runtime.h>
#include <hip/hip_bf16.h>
#include <math.h>

typedef __attribute__((ext_vector_type(2))) float v2f;
typedef __attribute__((ext_vector_type(8))) float v8f;

#define NB 16
#define NC 20
#define NH 256
#define NW 256
#define SLICE_ELEMS (NH * NW)           // 65536
#define NSLICE (NB * NC)                // 320
#define CHUNKS_PER_SLICE 4
#define CHUNK_ELEMS (SLICE_ELEMS / CHUNKS_PER_SLICE)  // 16384
#define THREADS 256
#define ITERS (CHUNK_ELEMS / (THREADS * 4))           // 16
#define NQ 10
#define TOTAL_N ((float)(NB * NC * NH * NW))          // 20971520

// Wave32 WMMA-based sum of 32 per-lane f32 values (full fp32 precision).
// A holds each lane's value once (other slot zero), B = all ones =>
// D row-sums partition the 64 A slots; every lane's 8 C/D VGPRs cover 8 of
// the 16 rows (lanes 0-15: M=0..7, lanes 16-31: M=8..15
// layout), so s + shfl_xor(s,16) = exact sum over all 32 lanes.
__device__ __forceinline__ float wave_sum_wmma(float v) {
    v2f a; a[0] = v;   a[1] = 0.0f;
    v2f b; b[0] = 1.0f; b[1] = 1.0f;
    v8f c = {};
    v8f d = __builtin_amdgcn_wmma_f32_16x16x4_f32(
        /*neg_a=*/false, a, /*neg_b=*/false, b,
        /*c_mod=*/(short)0, c, /*reuse_a=*/false, /*reuse_b=*/false);
    float s = d[0] + d[1] + d[2] + d[3] + d[4] + d[5] + d[6] + d[7];
    s += __shfl_xor(s, 16, 32);
    return s;
}

__global__ void __launch_bounds__(THREADS)
loss_partials(const float* __restrict__ pred,
              const float* __restrict__ tgt,
              float* __restrict__ ws) {
    const int slice = blockIdx.y;   // 0..319  -> (b,c)
    const int chunk = blockIdx.x;   // 0..3
    const int tid   = threadIdx.x;

    const size_t base = (size_t)slice * SLICE_ELEMS + (size_t)chunk * CHUNK_ELEMS;
    const float* __restrict__ ps = pred + base;
    const float* __restrict__ ts = tgt + base;
    const int pixbase = chunk * CHUNK_ELEMS;
    const int lin = tid * 4;

    // Coordinate math hoisted out of the stream loop:
    //   pix = pixbase + j*1024 + lin + e ; 1024 and pixbase are multiples of 256
    //   => x(pix) = (lin+e) & 255   (constant per thread/e)
    //   => y(pix) = (pixbase>>8) + 4*j + (lin>>8)   (uniform over e)
    float fx[4], fxsq[4];
#pragma unroll
    for (int e = 0; e < 4; ++e) {
        fx[e]   = (float)((lin + e) & 255);
        fxsq[e] = fx[e] * fx[e];
    }
    const float ybase = (float)((pixbase >> 8) + (lin >> 8));

    float acc[NQ];
#pragma unroll
    for (int q = 0; q < NQ; ++q) acc[q] = 0.0f;

#pragma unroll 4
    for (int j = 0; j < ITERS; ++j) {
        const int off = j * (THREADS * 4) + lin;
        const float4 pv = *(const float4*)(ps + off);
        const float4 tv = *(const float4*)(ts + off);
        const float px[4] = {pv.x, pv.y, pv.z, pv.w};
        const float tx[4] = {tv.x, tv.y, tv.z, tv.w};

        const float fy  = ybase + (float)(4 * j);
        const float fy2 = fy * fy;
        float psum = 0.0f, tsum = 0.0f;

#pragma unroll
        for (int e = 0; e < 4; ++e) {
            const float x = px[e];
            const float t = tx[e];
            // sigmoid via raw v_rcp (1-ulp), avoids IEEE div sequence
            const float en = __expf(-x);
            const float s  = 1.0f + en;
            const float p  = __builtin_amdgcn_rcpf(s);
            // focal: -log(pt+1e-8) ~= L (pos) or x + L (neg), L = log(1+e^-x)
            const float L   = __logf(s);
            const bool  pos = (t > 0.5f);
            const float om  = pos ? (1.0f - p) : p;   // 1 - pt
            const float nlp = pos ? L : (x + L);      // -log(pt)
            const float at  = pos ? 0.25f : 0.75f;
            acc[0] = fmaf(at * om * om, nlp, acc[0]);
            // sparsity
            const float dd = x - t;
            acc[1] = fmaf(dd, dd, acc[1]);
            acc[2] += fabsf(x);
            // per-slice moments (x-direction; y factored out below)
            tsum += t;
            psum += p;
            acc[5] = fmaf(t, fx[e],   acc[5]);
            acc[8] = fmaf(p, fx[e],   acc[8]);
            acc[9] = fmaf(p, fxsq[e], acc[9]);
        }
        acc[3] += tsum;
        acc[4] = fmaf(fy,  tsum, acc[4]);
        acc[6] += psum;
        acc[7] = fmaf(fy,  psum, acc[7]);
        acc[9] = fmaf(fy2, psum, acc[9]);
    }

    __shared__ float parts[NQ][THREADS];
#pragma unroll
    for (int q = 0; q < NQ; ++q) parts[q][tid] = acc[q];
    __syncthreads();

    if (tid < 32) {   // wave 0, all 32 lanes active -> EXEC all-1s for WMMA
        const int lane = tid;
        float* __restrict__ wrow = ws + (size_t)(slice * CHUNKS_PER_SLICE + chunk) * NQ;
        for (int q = 0; q < NQ; ++q) {
            float v = 0.0f;
#pragma unroll
            for (int i = 0; i < 8; ++i) v += parts[q][lane * 8 + i];
            const float s = wave_sum_wmma(v);
            if (lane == 0) wrow[q] = s;   // plain store: deterministic
        }
    }
}

__global__ void __launch_bounds__(512)
loss_finalize(const float* __restrict__ ws, float* __restrict__ out) {
    __shared__ float red[5][512];
    const int t = threadIdx.x;

    float focal = 0.0f, mse = 0.0f, l1 = 0.0f, conc = 0.0f, nval = 0.0f;
    if (t < NSLICE) {
        float q[NQ];
#pragma unroll
        for (int j = 0; j < NQ; ++j) q[j] = 0.0f;
        for (int c = 0; c < CHUNKS_PER_SLICE; ++c) {
            const float* __restrict__ row = ws + (size_t)(t * CHUNKS_PER_SLICE + c) * NQ;
#pragma unroll
            for (int j = 0; j < NQ; ++j) q[j] += row[j];
        }
        focal = q[0]; mse = q[1]; l1 = q[2];
        const float mass = q[3];
        const bool valid = (mass > 0.0f);
        const float safe = valid ? mass : 1.0f;
        const float cy = q[4] / safe;
        const float cx = q[5] / safe;
        // sum_p_dsq = Sp(y^2+x^2) - 2cy*Spy - 2cx*Spx + (cy^2+cx^2)*Sp
        const float per = (q[9] - 2.0f * cy * q[7] - 2.0f * cx * q[8]
                           + (cy * cy + cx * cx) * q[6]) * (1.0f / (float)SLICE_ELEMS);
        conc = valid ? per : 0.0f;
        nval = valid ? 1.0f : 0.0f;
    }
    red[0][t] = focal; red[1][t] = mse; red[2][t] = l1; red[3][t] = conc; red[4][t] = nval;
    __syncthreads();

    for (int s = 256; s > 0; s >>= 1) {
        if (t < s) {
#pragma unroll
            for (int j = 0; j < 5; ++j) red[j][t] += red[j][t + s];
        }
        __syncthreads();
    }

    if (t == 0) {
        const float invN = 1.0f / TOTAL_N;
        const float f  = red[0][0] * invN;
        const float sp = red[1][0] * invN + 1.0f /*SPARSITY_PENALTY*/ * red[2][0] * invN;
        const float nv = red[4][0];
        const float cc = (nv > 0.0f) ? (red[3][0] / nv) : 0.0f;   // CONC_INNER_W = 1.0
        const float total = 1.0f * f + 0.8f * sp + 1.5f * cc;
        out[0] = total;
        out[1] = f;
        out[2] = sp;
        out[3] = cc;
    }
}

extern "C" void kernel_launch(void* const* d_in, const int* in_sizes, int n_in,
                              void* d_out, int out_size, void* d_ws, size_t ws_size,
                              hipStream_t stream) {
    (void)in_sizes; (void)n_in; (void)out_size; (void)ws_size;
    const float* pred = (const float*)d_in[0];
    const float* tgt  = (const float*)d_in[1];
    float* ws  = (float*)d_ws;      // NSLICE*CHUNKS_PER_SLICE*NQ = 12800 floats
    float* out = (float*)d_out;     // 4 floats: total, focal, sparsity, concentration

    dim3 grid(CHUNKS_PER_SLICE, NSLICE);
    loss_partials<<<grid, THREADS, 0, stream>>>(pred, tgt, ws);
    loss_finalize<<<1, 512, 0, stream>>>(ws, out);
}